// ConvolutionalLUTLayer_13554916786204
// MI455X (gfx1250) — compile-verified
//
#include <hip/hip_runtime.h>
#include <cstdint>

// DWN convolutional LUT layer for MI455X (gfx1250, wave32).
// B=16, C=32, H=W=32, k=3 -> Hp=Wp=30, P=900, F=288, O=64.
// Layer sizes: 36 nodes -> 6 nodes -> 1 node, each a 6-input soft LUT.

#define TPB    128   // threads per block (4 wave32)
#define OG     8     // output channels handled per block
#define XROWS  8     // max image rows spanned by a 128-pixel tile (+2 halo)
#define XC     32    // channels
#define WP     30    // output width
#define PNUM   900   // pixels per image (30*30)
#define ONUM   64    // output channels
#define IMW    32    // input width
#define IMH    32

// Soft 6-input LUT evaluation by contracting one variable at a time:
// T_k[a] = T_{k+1}[a] + g_k * (T_{k+1}[a + 2^k] - T_{k+1}[a])
// L (64 floats, LDS) holds sigmoid(lut); bit i of address a selects g[i].
__device__ __forceinline__ float lut6_eval(const float* __restrict__ L,
                                           const float g[6]) {
    const float4* L4 = (const float4*)L;   // 256B-aligned rows in LDS
    float T[32];
    // contract bit 5: pairs (a, a+32); in float4 blocks: (blk, blk+8)
    #pragma unroll
    for (int a = 0; a < 8; ++a) {
        float4 lo = L4[a];
        float4 hi = L4[a + 8];
        T[4*a + 0] = fmaf(g[5], hi.x - lo.x, lo.x);
        T[4*a + 1] = fmaf(g[5], hi.y - lo.y, lo.y);
        T[4*a + 2] = fmaf(g[5], hi.z - lo.z, lo.z);
        T[4*a + 3] = fmaf(g[5], hi.w - lo.w, lo.w);
    }
    #pragma unroll
    for (int a = 0; a < 16; ++a) T[a] = fmaf(g[4], T[a + 16] - T[a], T[a]);
    #pragma unroll
    for (int a = 0; a < 8;  ++a) T[a] = fmaf(g[3], T[a + 8]  - T[a], T[a]);
    #pragma unroll
    for (int a = 0; a < 4;  ++a) T[a] = fmaf(g[2], T[a + 4]  - T[a], T[a]);
    #pragma unroll
    for (int a = 0; a < 2;  ++a) T[a] = fmaf(g[1], T[a + 2]  - T[a], T[a]);
    return fmaf(g[0], T[1] - T[0], T[0]);
}

__device__ __forceinline__ float sigmoidf_fast(float v) {
    return 1.0f / (1.0f + __expf(-v));
}

__global__ __launch_bounds__(TPB) void dwn_lut_kernel(
    const float* __restrict__ x,
    const int*   __restrict__ idx0,   // [64,36,6]
    const int*   __restrict__ idx1,   // [64,6,6]
    const int*   __restrict__ idx2,   // [64,1,6]
    const float* __restrict__ lut0,   // [64,36,64]
    const float* __restrict__ lut1,   // [64,6,64]
    const float* __restrict__ lut2,   // [64,1,64]
    float*       __restrict__ out)    // [16,64,30,30]
{
    // ---- LDS (~62 KB total) ----
    __shared__ float xwin[XC * XROWS * IMW];   // 32 KB staged x window
    __shared__ float h1s[36 * TPB];            // 18 KB layer-0 outputs, [s][tid]
    __shared__ float sl0[36 * 64];             // 9 KB  sigmoid(lut0[o])
    __shared__ float sl1[6 * 64];              // 1.5 KB
    __shared__ float sl2[64];
    __shared__ int   i0s[36 * 6];              // precomputed xwin offsets
    __shared__ int   i1s[6 * 6];               // precomputed h1s offsets
    __shared__ int   i2s[6];

    const int tid  = threadIdx.x;
    const int bi   = blockIdx.x >> 3;          // image index 0..15
    const int tile = blockIdx.x & 7;           // pixel tile 0..7
    const int p0   = tile * TPB;
    const int pend = (p0 + TPB < PNUM) ? (p0 + TPB) : PNUM;
    const int row0 = p0 / WP;
    const int nrows = (pend - 1) / WP + 3 - row0;   // rows needed (<= XROWS)

    // ---- stage x window into LDS with gfx1250 async copies (ASYNCcnt) ----
    // Copy nrows image rows of all 32 channels; per row: 32ch * 8 b128 units.
    {
        const uint64_t sbase   = (uint64_t)(uintptr_t)(x + (size_t)bi * (XC * IMH * IMW));
        const uint32_t ldsbase = (uint32_t)(uintptr_t)(&xwin[0]); // LDS byte offset
        for (int rrow = 0; rrow < nrows; ++rrow) {
            #pragma unroll
            for (int e = 0; e < 2; ++e) {          // 256 units / 128 threads
                const int u8 = tid + e * TPB;      // unit id within this row
                const int c  = u8 >> 3;            // channel
                const int u  = u8 & 7;             // 16B unit within row
                const uint32_t loff = ldsbase +
                    (uint32_t)(c * (XROWS * IMW * 4) + (rrow * 8 + u) * 16);
                const uint32_t goff =
                    (uint32_t)(c * (IMH * IMW * 4) + (row0 + rrow) * (IMW * 4) + u * 16);
                asm volatile("global_load_async_to_lds_b128 %0, %1, %2"
                             :: "v"(loff), "v"(goff), "s"(sbase)
                             : "memory");
            }
        }
    }

    // ---- per-thread pixel coordinates ----
    const int  p     = p0 + tid;
    const bool valid = (p < PNUM);
    const int  ph    = valid ? (p / WP) : row0;
    const int  pw    = valid ? (p - (p / WP) * WP) : 0;
    const int  gbase = (ph - row0) * IMW + pw;   // per-thread base into xwin

    const int obase = blockIdx.y * OG;

    for (int oo = 0; oo < OG; ++oo) {
        const int o = obase + oo;

        // ---- stage sigmoid(lut) slices + precomputed index offsets ----
        for (int e = tid; e < 36 * 64; e += TPB) sl0[e] = sigmoidf_fast(lut0[o * (36 * 64) + e]);
        for (int e = tid; e < 6 * 64;  e += TPB) sl1[e] = sigmoidf_fast(lut1[o * (6 * 64) + e]);
        if (tid < 64) sl2[tid] = sigmoidf_fast(lut2[o * 64 + tid]);
        for (int e = tid; e < 36 * 6; e += TPB) {
            // decode feature f = c*9 + di*3 + dj once; store final xwin offset
            const int f   = idx0[o * (36 * 6) + e];
            const int c   = f / 9;
            const int rem = f - 9 * c;
            const int di  = rem / 3;
            const int dj  = rem - 3 * di;
            i0s[e] = (c * XROWS + di) * IMW + dj;
        }
        if (tid < 36) i1s[tid] = idx1[o * 36 + tid] * TPB;
        if (tid < 6)  i2s[tid] = idx2[o * 6 + tid];

        if (oo == 0) {
            // x-window async copies must land before first compute
            asm volatile("s_wait_asynccnt 0x0" ::: "memory");
        }
        __syncthreads();

        // ---- layer 0: 36 nodes, gather from staged x window ----
        for (int s = 0; s < 36; ++s) {
            float g[6];
            #pragma unroll
            for (int i = 0; i < 6; ++i)
                g[i] = xwin[i0s[s * 6 + i] + gbase];   // one-add gather
            h1s[s * TPB + tid] = lut6_eval(&sl0[s * 64], g);
        }
        // h1s columns are lane-private: no barrier needed.

        // ---- layer 1: 6 nodes, gather from h1s ----
        float h2v[6];
        #pragma unroll
        for (int s = 0; s < 6; ++s) {
            float g[6];
            #pragma unroll
            for (int i = 0; i < 6; ++i)
                g[i] = h1s[i1s[s * 6 + i] + tid];
            h2v[s] = lut6_eval(&sl1[s * 64], g);
        }

        // ---- layer 2: 1 node, gather via 6-way select chain ----
        float g[6];
        #pragma unroll
        for (int i = 0; i < 6; ++i) {
            const int j = i2s[i];                   // wave-uniform, 0..5
            float v = h2v[0];
            v = (j == 1) ? h2v[1] : v;
            v = (j == 2) ? h2v[2] : v;
            v = (j == 3) ? h2v[3] : v;
            v = (j == 4) ? h2v[4] : v;
            v = (j == 5) ? h2v[5] : v;
            g[i] = v;
        }
        const float r = lut6_eval(sl2, g);

        if (valid) out[((size_t)bi * ONUM + o) * PNUM + p] = r;

        __syncthreads();   // protect LUT/idx LDS buffers before next o overwrites
    }
}

extern "C" void kernel_launch(void* const* d_in, const int* in_sizes, int n_in,
                              void* d_out, int out_size, void* d_ws, size_t ws_size,
                              hipStream_t stream) {
    (void)in_sizes; (void)n_in; (void)out_size; (void)d_ws; (void)ws_size;
    const float* x    = (const float*)d_in[0];
    const int*   idx0 = (const int*)  d_in[1];
    const int*   idx1 = (const int*)  d_in[2];
    const int*   idx2 = (const int*)  d_in[3];
    const float* l0   = (const float*)d_in[4];
    const float* l1   = (const float*)d_in[5];
    const float* l2   = (const float*)d_in[6];
    float* out = (float*)d_out;

    dim3 grid(16 * 8, ONUM / OG);   // (image x pixel-tile, o-group)
    dwn_lut_kernel<<<grid, TPB, 0, stream>>>(x, idx0, idx1, idx2, l0, l1, l2, out);
}